// LSTM_53558242181257
// MI455X (gfx1250) — compile-verified
//
#include <hip/hip_runtime.h>

// ---------------------------------------------------------------------------
// 2-layer LSTM, B=32, T=2000, H=F=512 on gfx1250 (MI455X).
// Single persistent kernel, device-wide barrier per superstep, layers
// pipelined with a 1-step skew. Recurrent+input GEMM done with
// V_WMMA_F32_16X16X4_F32 (fp32 in / fp32 accum).
// ---------------------------------------------------------------------------

#define T_STEPS 2000
#define BATCH   32
#define HID     512
#define GATES   2048          // 4*HID
#define NBLOCKS 32            // 16 blocks per layer
#define NTHREADS 256          // 8 wave32s

typedef float v2f __attribute__((ext_vector_type(2)));
typedef float v8f __attribute__((ext_vector_type(8)));

// ws layout (float offsets)
#define WS_H1A    0
#define WS_H1B    16384
#define WS_H2A    32768
#define WS_H2B    49152
#define WS_C1     65536
#define WS_C2     81920
#define WS_BC0    98304       // combined bias layer0: w0_b + u0_b  [2048]
#define WS_BC1    100352      // combined bias layer1: w1_b + u1_b  [2048]
#define WS_BAR    102400      // 2 ints: arrive counter, generation

__device__ __forceinline__ float sigmoidf_(float x) {
    return 1.0f / (1.0f + __expf(-x));
}

__device__ __forceinline__ void grid_sync(int* cnt, int* gen) {
    __syncthreads();
    if (threadIdx.x == 0) {
        __threadfence();
        int g = __hip_atomic_load(gen, __ATOMIC_RELAXED, __HIP_MEMORY_SCOPE_AGENT);
        int a = __hip_atomic_fetch_add(cnt, 1, __ATOMIC_ACQ_REL, __HIP_MEMORY_SCOPE_AGENT);
        if (a == NBLOCKS - 1) {
            __hip_atomic_store(cnt, 0, __ATOMIC_RELAXED, __HIP_MEMORY_SCOPE_AGENT);
            __hip_atomic_fetch_add(gen, 1, __ATOMIC_ACQ_REL, __HIP_MEMORY_SCOPE_AGENT);
        } else {
            while (__hip_atomic_load(gen, __ATOMIC_ACQUIRE, __HIP_MEMORY_SCOPE_AGENT) == g) {
                __builtin_amdgcn_s_sleep(1);
            }
        }
    }
    __syncthreads();
}

__global__ void lstm_init_kernel(float* ws,
                                 const float* w0_b, const float* u0_b,
                                 const float* w1_b, const float* u1_b) {
    int i = blockIdx.x * blockDim.x + threadIdx.x;
    if (i < 6 * BATCH * HID) ws[i] = 0.0f;              // h1A,h1B,h2A,h2B,c1,c2
    if (i < GATES) {
        ws[WS_BC0 + i] = w0_b[i] + u0_b[i];
        ws[WS_BC1 + i] = w1_b[i] + u1_b[i];
    }
    if (i == 0) {
        int* bar = (int*)(ws + WS_BAR);
        bar[0] = 0;
        bar[1] = 0;
    }
}

__global__ __launch_bounds__(NTHREADS)
void lstm_persistent_kernel(const float* __restrict__ x,
                            const float* __restrict__ w0_w,
                            const float* __restrict__ u0_w,
                            const float* __restrict__ w1_w,
                            const float* __restrict__ u1_w,
                            float* __restrict__ ws,
                            float* __restrict__ out) {
    __shared__ float gl[BATCH][132];   // gates staging: [batch][4 gates * 32 cols] + pad

    float* h1A = ws + WS_H1A;
    float* h1B = ws + WS_H1B;
    float* h2A = ws + WS_H2A;
    float* h2B = ws + WS_H2B;
    int*   cnt = (int*)(ws + WS_BAR);
    int*   gen = cnt + 1;

    float* out_h2 = out;                                       // [B,T,H]
    float* out_hh = out + (size_t)BATCH * T_STEPS * HID;       // [2,B,H]
    float* out_cc = out_hh + 2 * BATCH * HID;                  // [2,B,H]

    const int  lane    = threadIdx.x & 31;
    const int  wv      = threadIdx.x >> 5;   // wave 0..7
    const int  gate    = wv >> 1;            // 0..3 (i,f,g,o)
    const int  halfsel = wv & 1;             // which 16-col half of this block's 32 cols
    const bool isL2    = (blockIdx.x >= 16);
    const int  j       = isL2 ? (blockIdx.x - 16) : blockIdx.x;  // hidden-col block 0..15
    const int  n0      = gate * HID + j * 32 + halfsel * 16;     // gate-column base (of 2048)
    const int  col     = lane & 15;
    const int  koff    = (lane >> 4) * 2;    // A/B fragment K sub-offset per ISA layout

    // B operand: weights are [4H, K] row-major == B^T, so per-lane rows:
    const float* Wp = (isL2 ? w1_w : w0_w) + (size_t)(n0 + col) * HID;
    const float* Up = (isL2 ? u1_w : u0_w) + (size_t)(n0 + col) * HID;
    const float* bc = ws + (isL2 ? WS_BC1 : WS_BC0);
    float*       cb = ws + (isL2 ? WS_C2 : WS_C1);

    for (int s = 0; s <= T_STEPS; ++s) {
        const bool active = isL2 ? (s >= 1) : (s < T_STEPS);
        const int  t      = isL2 ? (s - 1) : s;

        if (active) {
            v8f accLo = {};   // batches 0..15
            v8f accHi = {};   // batches 16..31

            // ---- segment 0: input projection (x_t for L1, h1[t] for L2) ----
            const float *aLo, *aHi;
            if (!isL2) {
                aLo = x + ((size_t)col        * T_STEPS + t) * HID;
                aHi = x + ((size_t)(col + 16) * T_STEPS + t) * HID;
            } else {
                const float* h1in = (t & 1) ? h1B : h1A;   // h1[t] lives in buf[t&1]
                aLo = h1in + col * HID;
                aHi = h1in + (col + 16) * HID;
            }
#pragma unroll 4
            for (int k = 0; k < HID; k += 4) {
                v2f a0 = *(const v2f*)(aLo + k + koff);
                v2f a1 = *(const v2f*)(aHi + k + koff);
                v2f bb = *(const v2f*)(Wp  + k + koff);
                accLo = __builtin_amdgcn_wmma_f32_16x16x4_f32(
                    false, a0, false, bb, (short)0, accLo, false, false);
                accHi = __builtin_amdgcn_wmma_f32_16x16x4_f32(
                    false, a1, false, bb, (short)0, accHi, false, false);
            }

            // ---- segment 1: recurrent projection (h_prev) ----
            const float* hp;
            if (!isL2) hp = (s & 1) ? h1A : h1B;   // h1[s-1] in buf[(s-1)&1]
            else       hp = (s & 1) ? h2B : h2A;   // h2[s-2] in buf[(s-2)&1]
            const float* bLo = hp + col * HID;
            const float* bHi = hp + (col + 16) * HID;
#pragma unroll 4
            for (int k = 0; k < HID; k += 4) {
                v2f a0 = *(const v2f*)(bLo + k + koff);
                v2f a1 = *(const v2f*)(bHi + k + koff);
                v2f bb = *(const v2f*)(Up  + k + koff);
                accLo = __builtin_amdgcn_wmma_f32_16x16x4_f32(
                    false, a0, false, bb, (short)0, accLo, false, false);
                accHi = __builtin_amdgcn_wmma_f32_16x16x4_f32(
                    false, a1, false, bb, (short)0, accHi, false, false);
            }

            // ---- spill accumulators to LDS (ISA C/D layout) ----
            const int gcol = gate * 32 + halfsel * 16 + col;
#pragma unroll
            for (int r = 0; r < 8; ++r) {
                const int brow = r + (lane >> 4) * 8;
                gl[brow][gcol]      = accLo[r];
                gl[brow + 16][gcol] = accHi[r];
            }
        }
        __syncthreads();

        if (active) {
            // ---- elementwise LSTM update for this block's 32 hidden cols ----
            float* hw;
            if (!isL2) hw = (s & 1) ? h1B : h1A;   // write h1[s]   -> buf[s&1]
            else       hw = (s & 1) ? h2A : h2B;   // write h2[s-1] -> buf[(s-1)&1]

            for (int e = threadIdx.x; e < BATCH * 32; e += NTHREADS) {
                const int b  = e >> 5;
                const int nl = e & 31;
                const int n  = j * 32 + nl;
                const float iv = gl[b][nl]      + bc[n];
                const float fv = gl[b][32 + nl] + bc[HID + n];
                const float gv = gl[b][64 + nl] + bc[2 * HID + n];
                const float ov = gl[b][96 + nl] + bc[3 * HID + n];
                const float ig = sigmoidf_(iv);
                const float fg = sigmoidf_(fv);
                const float gg = tanhf(gv);
                const float og = sigmoidf_(ov);
                const float co = cb[b * HID + n];
                const float cn = fg * co + ig * gg;
                cb[b * HID + n] = cn;
                const float hn = og * tanhf(cn);
                hw[b * HID + n] = hn;
                if (isL2) out_h2[((size_t)b * T_STEPS + t) * HID + n] = hn;
                if (t == T_STEPS - 1) {
                    const int l = isL2 ? 1 : 0;
                    out_hh[(l * BATCH + b) * HID + n] = hn;
                    out_cc[(l * BATCH + b) * HID + n] = cn;
                }
            }
        }
        grid_sync(cnt, gen);
    }
}

extern "C" void kernel_launch(void* const* d_in, const int* in_sizes, int n_in,
                              void* d_out, int out_size, void* d_ws, size_t ws_size,
                              hipStream_t stream) {
    const float* x   = (const float*)d_in[0];
    const float* w0w = (const float*)d_in[1];
    const float* w0b = (const float*)d_in[2];
    const float* u0w = (const float*)d_in[3];
    const float* u0b = (const float*)d_in[4];
    const float* w1w = (const float*)d_in[5];
    const float* w1b = (const float*)d_in[6];
    const float* u1w = (const float*)d_in[7];
    const float* u1b = (const float*)d_in[8];
    float* out = (float*)d_out;
    float* wsf = (float*)d_ws;

    lstm_init_kernel<<<(6 * BATCH * HID + 255) / 256, 256, 0, stream>>>(
        wsf, w0b, u0b, w1b, u1b);
    lstm_persistent_kernel<<<NBLOCKS, NTHREADS, 0, stream>>>(
        x, w0w, u0w, w1w, u1w, wsf, out);
}